// GATLayer_34660386078860
// MI455X (gfx1250) — compile-verified
//
#include <hip/hip_runtime.h>
#include <math.h>

#define IN_DIM  128
#define OUT_DIM 128

typedef __attribute__((ext_vector_type(2))) float v2f;
typedef __attribute__((ext_vector_type(8))) float v8f;

// ---- order-preserving float <-> uint mapping (max(float) == max(uint key)) ----
__device__ __forceinline__ unsigned f2ord(float f) {
  unsigned b = __float_as_uint(f);
  return (b & 0x80000000u) ? ~b : (b | 0x80000000u);
}
__device__ __forceinline__ float ord2f(unsigned o) {
  unsigned b = (o & 0x80000000u) ? (o & 0x7FFFFFFFu) : ~o;
  return __uint_as_float(b);
}

// ---- K0: zero output, init segment-max keys and denominators ----
__global__ void init_kernel(float4* __restrict__ out4, unsigned* __restrict__ m_key,
                            float* __restrict__ denom, int n4, int nNodes) {
  int i = blockIdx.x * blockDim.x + threadIdx.x;
  if (i < n4)     out4[i] = make_float4(0.f, 0.f, 0.f, 0.f);
  if (i < nNodes) { m_key[i] = 0u; denom[i] = 0.f; }
}

// ---- K1: z[N,128] = h[N,128] @ W[128,128]^T via V_WMMA_F32_16X16X4_F32 ----
// One wave computes one 16x16 tile of z; 8 waves/block = 16 rows x all 128 cols.
// A(16x4 f32): lane l<16 holds {K0,K1} of row l; lanes 16..31 hold {K2,K3}.
// B(4x16 f32): same pattern with lanes = output columns; B[k][n] = W[n][k].
// D(16x16 f32): VGPR r -> row r (lanes 0-15) / row r+8 (lanes 16-31).
__global__ __launch_bounds__(256) void gemm_wmma(const float* __restrict__ h,
                                                 const float* __restrict__ W,
                                                 float* __restrict__ z, int nNodes) {
  const int wave = threadIdx.x >> 5;
  const int lane = threadIdx.x & 31;
  const int half = lane >> 4;   // 0: K0/K1 pair, 1: K2/K3 pair
  const int l    = lane & 15;
  const int row0 = blockIdx.x << 4;
  const int col0 = wave << 4;

  const int arow = min(row0 + l, nNodes - 1);          // clamp loads, keep EXEC full
  const float2* ap = (const float2*)(h + (size_t)arow * IN_DIM + 2 * half);
  const float2* bp = (const float2*)(W + (size_t)(col0 + l) * IN_DIM + 2 * half);

  v8f acc = {};
#pragma unroll
  for (int kk = 0; kk < IN_DIM / 4; ++kk) {            // 32 chained WMMAs, K=4 each
    float2 av = ap[2 * kk];
    float2 bv = bp[2 * kk];
    v2f a; a.x = av.x; a.y = av.y;
    v2f b; b.x = bv.x; b.y = bv.y;
    acc = __builtin_amdgcn_wmma_f32_16x16x4_f32(
        /*neg_a=*/false, a, /*neg_b=*/false, b,
        /*c_mod=*/(short)0, acc, /*reuse_a=*/false, /*reuse_b=*/false);
  }

  const int rbase = row0 + 8 * half;
  float* zp = z + (size_t)rbase * OUT_DIM + col0 + l;
#pragma unroll
  for (int r = 0; r < 8; ++r)
    if (rbase + r < nNodes) zp[(size_t)r * OUT_DIM] = acc[r];
}

// ---- K2: per-edge RBF score e = -beta*||z_s - z_d||, segment-max via uint atomicMax ----
// One wave per edge; lane holds 4 consecutive channels (float4 gather, L2-resident z).
__global__ __launch_bounds__(256) void edge_score(const float* __restrict__ z,
                                                  const int* __restrict__ src,
                                                  const int* __restrict__ dst,
                                                  const float* __restrict__ beta_p,
                                                  float* __restrict__ e,
                                                  unsigned* __restrict__ m_key, int nE) {
  int w    = (int)((blockIdx.x * 256u + threadIdx.x) >> 5);
  int lane = threadIdx.x & 31;
  if (w >= nE) return;
  int s = src[w], d = dst[w];
  float4 a = ((const float4*)(z + (size_t)s * OUT_DIM))[lane];
  float4 b = ((const float4*)(z + (size_t)d * OUT_DIM))[lane];
  float dx = a.x - b.x, dy = a.y - b.y, dz = a.z - b.z, dw = a.w - b.w;
  float sq = dx * dx + dy * dy + dz * dz + dw * dw;
#pragma unroll
  for (int off = 16; off > 0; off >>= 1) sq += __shfl_xor(sq, off, 32);
  if (lane == 0) {
    float ev = -beta_p[0] * sqrtf(sq + 1e-12f);
    e[w] = ev;
    atomicMax(&m_key[d], f2ord(ev));
  }
}

// ---- K3: ex = exp(e - m[dst]); denom[dst] += ex (thread per edge) ----
__global__ void edge_exp(const int* __restrict__ dst, const unsigned* __restrict__ m_key,
                         float* __restrict__ e, float* __restrict__ denom, int nE) {
  int i = blockIdx.x * blockDim.x + threadIdx.x;
  if (i >= nE) return;
  int d = dst[i];
  float ex = __expf(e[i] - ord2f(m_key[d]));
  e[i] = ex;
  atomicAdd(&denom[d], ex);
}

// ---- K4: h_out[dst] += (ex/denom[dst]) * z[src]  (wave per edge, f32 L2 atomics) ----
__global__ __launch_bounds__(256) void edge_scatter(const float* __restrict__ z,
                                                    const int* __restrict__ src,
                                                    const int* __restrict__ dst,
                                                    const float* __restrict__ ex,
                                                    const float* __restrict__ denom,
                                                    float* __restrict__ out, int nE) {
  int w    = (int)((blockIdx.x * 256u + threadIdx.x) >> 5);
  int lane = threadIdx.x & 31;
  if (w >= nE) return;
  int s = src[w], d = dst[w];
  float alpha = ex[w] / denom[d];
  float4 v = ((const float4*)(z + (size_t)s * OUT_DIM))[lane];
  float* o = out + (size_t)d * OUT_DIM + lane * 4;
  atomicAdd(o + 0, alpha * v.x);
  atomicAdd(o + 1, alpha * v.y);
  atomicAdd(o + 2, alpha * v.z);
  atomicAdd(o + 3, alpha * v.w);
}

extern "C" void kernel_launch(void* const* d_in, const int* in_sizes, int n_in,
                              void* d_out, int out_size, void* d_ws, size_t ws_size,
                              hipStream_t stream) {
  const float* h    = (const float*)d_in[0];
  const float* Wfc  = (const float*)d_in[1];
  const float* beta = (const float*)d_in[2];
  const int*   src  = (const int*)d_in[3];
  const int*   dst  = (const int*)d_in[4];
  float* out = (float*)d_out;

  const int nNodes = in_sizes[0] / IN_DIM;
  const int nE     = in_sizes[3];

  // workspace layout (256B-aligned slices)
  char* ws = (char*)d_ws;
  float* z = (float*)ws;
  size_t zB = ((size_t)nNodes * OUT_DIM * sizeof(float) + 255) & ~(size_t)255;
  float* e = (float*)(ws + zB);
  size_t eB = ((size_t)nE * sizeof(float) + 255) & ~(size_t)255;
  unsigned* m_key = (unsigned*)(ws + zB + eB);
  size_t mB = ((size_t)nNodes * sizeof(unsigned) + 255) & ~(size_t)255;
  float* denom = (float*)(ws + zB + eB + mB);

  const int n4 = nNodes * OUT_DIM / 4;
  init_kernel<<<(n4 + 255) / 256, 256, 0, stream>>>((float4*)out, m_key, denom, n4, nNodes);

  const int rowTiles = (nNodes + 15) / 16;
  gemm_wmma<<<rowTiles, 256, 0, stream>>>(h, Wfc, z, nNodes);

  const int waveBlocks = (nE + 7) / 8;   // 8 waves (edges) per 256-thread block
  edge_score<<<waveBlocks, 256, 0, stream>>>(z, src, dst, beta, e, m_key, nE);
  edge_exp<<<(nE + 255) / 256, 256, 0, stream>>>(dst, m_key, e, denom, nE);
  edge_scatter<<<waveBlocks, 256, 0, stream>>>(z, src, dst, e, denom, out, nE);
}